// ResNetVQC_12936441496147
// MI455X (gfx1250) — compile-verified
//
#include <hip/hip_runtime.h>

typedef __attribute__((ext_vector_type(2))) float v2f;
typedef __attribute__((ext_vector_type(8))) float v8f;

#define NBLK    32
#define NLAYERS 6
#define QDEPTH  6
#define TILE    16
#define WAVES_PER_BLK 4
#define NFEAT   96
#define HSTRIDE (NFEAT * TILE)   // 1536 floats per H buffer

// ---------------------------------------------------------------------------
// Kernel 1: compose per-(layer,block) 3-qubit circuit into a real 16x8 matrix
//   R[l*32+k] : rows 0..7 = Re(C8), rows 8..15 = Im(C8); psi_in is real.
// ---------------------------------------------------------------------------
__global__ void vqc_build_mats(const float* __restrict__ thetas,
                               float* __restrict__ R) {
  int id = blockIdx.x * blockDim.x + threadIdx.x;
  if (id >= NLAYERS * NBLK) return;

  float re[8][8], im[8][8];
  for (int i = 0; i < 8; ++i)
    for (int j = 0; j < 8; ++j) { re[i][j] = (i == j) ? 1.0f : 0.0f; im[i][j] = 0.0f; }

  const float* W = thetas + (size_t)id * (QDEPTH * 9);  // [d][w][3]
  for (int d = 0; d < QDEPTH; ++d) {
    for (int w = 0; w < 3; ++w) {
      float phi = W[(d * 3 + w) * 3 + 0];
      float th  = W[(d * 3 + w) * 3 + 1];
      float om  = W[(d * 3 + w) * 3 + 2];
      float ct = __cosf(0.5f * th), st = __sinf(0.5f * th);
      float a = 0.5f * (phi + om), b = 0.5f * (phi - om);
      float ca = __cosf(a), sa = __sinf(a);
      float cb = __cosf(b), sb = __sinf(b);
      // m00 = e^{-ia}ct, m01 = -e^{ib}st, m10 = e^{-ib}st, m11 = e^{ia}ct
      float m00r =  ca * ct, m00i = -sa * ct;
      float m01r = -cb * st, m01i = -sb * st;
      float m10r =  cb * st, m10i = -sb * st;
      float m11r =  ca * ct, m11i =  sa * ct;
      int sh = 2 - w;
      for (int i = 0; i < 8; ++i) {
        if ((i >> sh) & 1) continue;
        int i1 = i | (1 << sh);
        for (int c = 0; c < 8; ++c) {
          float r0 = re[i][c],  i0 = im[i][c];
          float r1 = re[i1][c], i1v = im[i1][c];
          re[i][c]  = m00r * r0 - m00i * i0 + m01r * r1 - m01i * i1v;
          im[i][c]  = m00r * i0 + m00i * r0 + m01r * i1v + m01i * r1;
          re[i1][c] = m10r * r0 - m10i * i0 + m11r * r1 - m11i * i1v;
          im[i1][c] = m10r * i0 + m10i * r0 + m11r * i1v + m11i * r1;
        }
      }
    }
    int rng = (d % 2) + 1;
    for (int w = 0; w < 3; ++w) {
      int t = (w + rng) % 3;
      int cbit = 1 << (2 - w), tbit = 1 << (2 - t);
      for (int i = 0; i < 8; ++i) {
        if (!(i & cbit) || (i & tbit)) continue;
        int i1 = i | tbit;  // swap rows i <-> i1 (left-multiply by CNOT)
        for (int c = 0; c < 8; ++c) {
          float tr = re[i][c]; re[i][c] = re[i1][c]; re[i1][c] = tr;
          float ti = im[i][c]; im[i][c] = im[i1][c]; im[i1][c] = ti;
        }
      }
    }
  }
  float* o = R + (size_t)id * 128;  // [16][8] row-major
  for (int m = 0; m < 8; ++m)
    for (int c = 0; c < 8; ++c) {
      o[m * 8 + c]       = re[m][c];
      o[(m + 8) * 8 + c] = im[m][c];
    }
}

// ---------------------------------------------------------------------------
// One quantum layer over all 32 blocks for this wave's 16-sample tile.
// MODE: 0 = stem (contiguous, no residual)
//       1 = residual ('fully' permuted inputs, residual add)
//       2 = reduce (contiguous, <Z_0> only)
// All mode decisions are compile-time; write-back is unguarded (both lane
// halves hold identical z values after the cross-half combine).
// ---------------------------------------------------------------------------
template <int MODE>
__device__ __forceinline__ void vqc_layer(const float* __restrict__ Hc,
                                          float* __restrict__ Hn,
                                          const float* __restrict__ Rbuf,
                                          int n, int half) {
#pragma unroll 2
  for (int k = 0; k < NBLK; ++k) {
    const int p0 = 3 * k, p1 = 3 * k + 1, p2 = 3 * k + 2;
    const int f0 = (MODE == 1) ? (3 * (p0 & 31) + (p0 >> 5)) : p0;
    const int f1 = (MODE == 1) ? (3 * (p1 & 31) + (p1 >> 5)) : p1;
    const int f2 = (MODE == 1) ? (3 * (p2 & 31) + (p2 >> 5)) : p2;
    float a0v = Hc[f0 * TILE + n];
    float a1v = Hc[f1 * TILE + n];
    float a2v = Hc[f2 * TILE + n];

    float c0 = __cosf(0.5f * a0v), sn0 = __sinf(0.5f * a0v);
    float c1 = __cosf(0.5f * a1v), sn1 = __sinf(0.5f * a1v);
    float c2 = __cosf(0.5f * a2v), sn2 = __sinf(0.5f * a2v);

    // B-fragment values for this lane: K = q0*4 + half*2 + v
    //   psi_K = t(q0) * t(half) * t(v),  t = (bit ? sin : cos)
    float T1 = half ? sn1 : c1;
    float u0 = c0 * T1, u1 = sn0 * T1;
    v2f b0, b1;
    b0.x = u0 * c2;  b0.y = u0 * sn2;      // q0=0 (K = 0..3 slice)
    b1.x = u1 * c2;  b1.y = u1 * sn2;      // q0=1 (K = 4..7 slice)

    // A-fragments from LDS (8-byte aligned -> ds_load_2addr_b64)
    const float* Rm = Rbuf + k * 128 + n * 8 + 2 * half;
    v2f a0 = *(const v2f*)(Rm);            // K = {2h, 2h+1}
    v2f a1 = *(const v2f*)(Rm + 4);        // K = {4+2h, 4+2h+1}

    v8f acc = {};
    acc = __builtin_amdgcn_wmma_f32_16x16x4_f32(false, a0, false, b0,
                                                (short)0, acc, false, false);
    acc = __builtin_amdgcn_wmma_f32_16x16x4_f32(false, a1, false, b1,
                                                (short)0, acc, false, false);

    // Signed partial sums per half (half0: Sum +-Re^2, half1: Sum +-Im^2),
    // then one cross-half combine per z component.
    float z0 = 0.f, z1 = 0.f, z2 = 0.f;
#pragma unroll
    for (int r = 0; r < 8; ++r) {
      float sq = acc[r] * acc[r];
      z0 += (r & 4) ? -sq : sq;            // qubit 0 sign
      z1 += (r & 2) ? -sq : sq;            // qubit 1 sign
      z2 += (r & 1) ? -sq : sq;            // qubit 2 sign
    }
    z0 += __shfl_xor(z0, 16, 32);
    if (MODE != 2) {
      z1 += __shfl_xor(z1, 16, 32);
      z2 += __shfl_xor(z2, 16, 32);
    }

    // Both halves hold identical values -> unguarded same-value stores.
    if (MODE == 2) {
      Hn[k * TILE + n] = z0;
    } else if (MODE == 1) {
      Hn[p0 * TILE + n] = z0 + Hc[p0 * TILE + n];
      Hn[p1 * TILE + n] = z1 + Hc[p1 * TILE + n];
      Hn[p2 * TILE + n] = z2 + Hc[p2 * TILE + n];
    } else {
      Hn[p0 * TILE + n] = z0;
      Hn[p1 * TILE + n] = z1;
      Hn[p2 * TILE + n] = z2;
    }
  }
}

// ---------------------------------------------------------------------------
// Kernel 2: main pipeline. One wave32 owns a 16-sample tile; per (layer,block)
// two V_WMMA_F32_16X16X4_F32 compute all 16 amplitude components x 16 samples.
// ---------------------------------------------------------------------------
__launch_bounds__(WAVES_PER_BLK * 32, 1)
__global__ void vqc_main(const float* __restrict__ x,
                         const float* __restrict__ R,
                         const float* __restrict__ w_cls,
                         const float* __restrict__ b_cls,
                         float* __restrict__ out) {
  // 4 waves * 2 H-buffers * 96*16 floats (48 KB) + 32*128 floats Rbuf (16 KB)
  __shared__ float shmem[WAVES_PER_BLK * 2 * HSTRIDE + NBLK * 128];

  const int tid  = threadIdx.x;
  const int lane = tid & 31;
  const int wave = tid >> 5;
  const int n    = lane & 15;   // sample within tile; also A-fragment row m
  const int half = lane >> 4;   // K-half selector for A/B fragments

  const int tileIdx = blockIdx.x * WAVES_PER_BLK + wave;
  const int sbase   = tileIdx * TILE;

  float* Hc   = shmem + wave * 2 * HSTRIDE;
  float* Hn   = Hc + HSTRIDE;
  float* Rbuf = shmem + WAVES_PER_BLK * 2 * HSTRIDE;

  // Load x tile (coalesced on feature axis): Hc[f][n] = x[(sbase+nn)*96 + f]
  for (int nn = 0; nn < TILE; ++nn) {
    const float* xr = x + (size_t)(sbase + nn) * NFEAT;
    for (int f = lane; f < NFEAT; f += 32)
      Hc[f * TILE + nn] = xr[f];
  }
  __syncthreads();

  // ---- stem (layer 0) ----
  for (int i = tid; i < NBLK * 128; i += WAVES_PER_BLK * 32)
    Rbuf[i] = R[i];
  __syncthreads();
  vqc_layer<0>(Hc, Hn, Rbuf, n, half);
  { float* t = Hc; Hc = Hn; Hn = t; }
  __syncthreads();

  // ---- residual layers 1..4 ----
  for (int l = 1; l <= 4; ++l) {
    const float* Rl = R + (size_t)l * NBLK * 128;
    for (int i = tid; i < NBLK * 128; i += WAVES_PER_BLK * 32)
      Rbuf[i] = Rl[i];
    __syncthreads();
    vqc_layer<1>(Hc, Hn, Rbuf, n, half);
    float* t = Hc; Hc = Hn; Hn = t;
    __syncthreads();
  }

  // ---- reduce layer (layer 5) ----
  {
    const float* Rl = R + (size_t)(NLAYERS - 1) * NBLK * 128;
    for (int i = tid; i < NBLK * 128; i += WAVES_PER_BLK * 32)
      Rbuf[i] = Rl[i];
    __syncthreads();
    vqc_layer<2>(Hc, Hn, Rbuf, n, half);
    float* t = Hc; Hc = Hn; Hn = t;
    __syncthreads();
  }

  // Classifier: z in Hc[k*16+n], out[b,c] = sum_k z_k * w_cls[c,k] + b_cls[c]
#pragma unroll
  for (int ci = 0; ci < 5; ++ci) {
    int c = half * 5 + ci;
    float acc = b_cls[c];
#pragma unroll
    for (int k = 0; k < NBLK; ++k)
      acc += Hc[k * TILE + n] * w_cls[c * NBLK + k];
    out[(size_t)(sbase + n) * 10 + c] = acc;
  }
}

// ---------------------------------------------------------------------------
extern "C" void kernel_launch(void* const* d_in, const int* in_sizes, int n_in,
                              void* d_out, int out_size, void* d_ws, size_t ws_size,
                              hipStream_t stream) {
  const float* x      = (const float*)d_in[0];  // (65536, 96)
  const float* thetas = (const float*)d_in[1];  // (6, 32, 6, 3, 3)
  const float* w_cls  = (const float*)d_in[2];  // (10, 32)
  const float* b_cls  = (const float*)d_in[3];  // (10,)
  float* out = (float*)d_out;                   // (65536, 10)
  float* R   = (float*)d_ws;                    // 192 * 128 floats = 96 KB

  vqc_build_mats<<<NLAYERS, NBLK, 0, stream>>>(thetas, R);

  const int B = 65536;
  const int samplesPerBlock = WAVES_PER_BLK * TILE;  // 64
  vqc_main<<<B / samplesPerBlock, WAVES_PER_BLK * 32, 0, stream>>>(
      x, R, w_cls, b_cls, out);
}